// Focus2D_63419487092925
// MI455X (gfx1250) — compile-verified
//
#include <hip/hip_runtime.h>
#include <stdint.h>

// Focus2D for MI455X (gfx1250, wave32).
// Bandwidth-bound (~300-500 MB traffic, ~0.5 GFLOP): no WMMA path is profitable.
// CDNA5-specific paths used:
//   - tensor_load_to_lds (TDM) + s_wait_tensorcnt: stage per-(b,c) params into LDS
//   - global_prefetch_b8 on the streaming scan
//   - wave32 __shfl_xor butterfly OR-reduction; channel-per-lane coalescing

static constexpr int B = 4, H = 512, W = 512, C = 32, PAD = 3;

struct alignas(16) Params {
    int   rh, rw;          // resized height/width
    int   sy_off, sx_off;  // ri = oi + sy_off ; rj = oj + sx_off
    int   ym, xm;          // crop origin
    int   hcm1, wcm1;      // hc-1, wc-1
    float fy, fx;          // hc/rh, wc/rw
    int   pad[6];          // pad to 64 B so 32 entries == 2048 B (TDM tile)
};
static_assert(sizeof(Params) == 64, "Params must be 64B");

#ifndef USE_TDM_PARAMS
#define USE_TDM_PARAMS 1
#endif
#if USE_TDM_PARAMS && __has_builtin(__builtin_amdgcn_tensor_load_to_lds)
#define HAVE_TDM 1
#else
#define HAVE_TDM 0
#endif

typedef uint32_t u32x4 __attribute__((ext_vector_type(4)));
typedef int      i32x4 __attribute__((ext_vector_type(4)));
typedef int      i32x8 __attribute__((ext_vector_type(8)));

// OR-reduce 256 uint32s to one (thread 0 of the block holds the result).
__device__ inline uint32_t block_or_256(uint32_t m) {
    __shared__ uint32_t sh[256];
    const int tid = threadIdx.x;
    sh[tid] = m;
    __syncthreads();
    if (tid < 128) sh[tid] |= sh[tid + 128];
    __syncthreads();
    if (tid < 64)  sh[tid] |= sh[tid + 64];
    __syncthreads();
    uint32_t v = 0;
    if (tid < 32) {                       // wave 0: wave32 butterfly
        v = sh[tid] | sh[tid + 32];
        v |= __shfl_xor(v, 16);
        v |= __shfl_xor(v, 8);
        v |= __shfl_xor(v, 4);
        v |= __shfl_xor(v, 2);
        v |= __shfl_xor(v, 1);
    }
    return v;
}

// ---------------------------------------------------------------------------
// Kernel 1: row_any[b*H+h] = bitmask over c of any(x[b,h,:,c] > 0)   (y_scan)
// grid = B*H blocks, 256 threads: channel-quad per lane, float4 loads (512B/wave).
// ---------------------------------------------------------------------------
__global__ void row_scan_kernel(const float* __restrict__ x,
                                uint32_t* __restrict__ row_any) {
    const int b  = blockIdx.x >> 9;         // H == 512
    const int h  = blockIdx.x & (H - 1);
    const int c4 = threadIdx.x & 7;         // channel quad 0..7 -> channels 4*c4..
    const int wq = threadIdx.x >> 3;        // 0..31
    const float4* row = (const float4*)(x + (((size_t)b * H + h) * W) * C);

    int a0 = 0, a1 = 0, a2 = 0, a3 = 0;
    for (int k = 0; k < 16; ++k) {
        const int w = wq + 32 * k;
        if (k + 4 < 16)
            __builtin_prefetch(row + (size_t)(w + 128) * 8 + c4, 0, 0); // global_prefetch_b8
        const float4 f = row[(size_t)w * 8 + c4];
        a0 |= (f.x > 0.0f); a1 |= (f.y > 0.0f);
        a2 |= (f.z > 0.0f); a3 |= (f.w > 0.0f);
    }
    const uint32_t m = (uint32_t)(a0 | (a1 << 1) | (a2 << 2) | (a3 << 3)) << (c4 * 4);
    const uint32_t r = block_or_256(m);
    if (threadIdx.x == 0) row_any[(size_t)b * H + h] = r;
}

// ---------------------------------------------------------------------------
// Kernel 2: col_any[b*W+w] = bitmask over c of any(x[b,:,w,c] > 0)   (x_scan)
// grid = B*W blocks. Second pass is L2-resident (134 MB < 192 MB L2).
// ---------------------------------------------------------------------------
__global__ void col_scan_kernel(const float* __restrict__ x,
                                uint32_t* __restrict__ col_any) {
    const int b  = blockIdx.x >> 9;         // W == 512
    const int w  = blockIdx.x & (W - 1);
    const int c4 = threadIdx.x & 7;
    const int hq = threadIdx.x >> 3;        // 0..31

    int a0 = 0, a1 = 0, a2 = 0, a3 = 0;
    for (int k = 0; k < 16; ++k) {
        const int h = hq + 32 * k;
        const float4 f = *(const float4*)(x + ((((size_t)b * H + h) * W + w) * C) + 4 * c4);
        a0 |= (f.x > 0.0f); a1 |= (f.y > 0.0f);
        a2 |= (f.z > 0.0f); a3 |= (f.w > 0.0f);
    }
    const uint32_t m = (uint32_t)(a0 | (a1 << 1) | (a2 << 2) | (a3 << 3)) << (c4 * 4);
    const uint32_t r = block_or_256(m);
    if (threadIdx.x == 0) col_any[(size_t)b * W + w] = r;
}

// ---------------------------------------------------------------------------
// Kernel 3: bbox + resize-constant precompute per (b,c).
// grid = B*C blocks of 512 threads (H == W == 512).
// argmax semantics: no active index -> min = max = 0 (matches jnp argmax of 0s).
// ---------------------------------------------------------------------------
__global__ void bbox_kernel(const uint32_t* __restrict__ row_any,
                            const uint32_t* __restrict__ col_any,
                            Params* __restrict__ params) {
    const int b = blockIdx.x >> 5;          // C == 32
    const int c = blockIdx.x & 31;
    __shared__ int s_ymin, s_ymax, s_xmin, s_xmax;
    if (threadIdx.x == 0) { s_ymin = 0x7fffffff; s_ymax = -1; s_xmin = 0x7fffffff; s_xmax = -1; }
    __syncthreads();

    const int i = threadIdx.x;
    if ((row_any[(size_t)b * H + i] >> c) & 1u) { atomicMin(&s_ymin, i); atomicMax(&s_ymax, i); }
    if ((col_any[(size_t)b * W + i] >> c) & 1u) { atomicMin(&s_xmin, i); atomicMax(&s_xmax, i); }
    __syncthreads();

    if (threadIdx.x == 0) {
        const int ym = (s_ymax < 0) ? 0 : s_ymin;
        const int yM = (s_ymax < 0) ? 0 : s_ymax;
        const int xm = (s_xmax < 0) ? 0 : s_xmin;
        const int xM = (s_xmax < 0) ? 0 : s_xmax;

        const int hc = max(yM - ym, 1);
        const int wc = max(xM - xm, 1);
        const int zh = max(H - 2 * PAD, hc);
        const int zw = max(W - 2 * PAD, wc);
        const float scale = fminf((float)zh / (float)hc, (float)zw / (float)wc);
        const int rh = __float2int_rn(scale * (float)hc);   // RNE == jnp.round
        const int rw = __float2int_rn(scale * (float)wc);

        // ri = oi - max(0,(H-rh)//2) + max(0,(rh-H)//2); floor-div == C-div after max(0, )
        const int pad_t  = max(0, (H - rh) / 2);
        const int crop_t = max(0, (rh - H) / 2);
        const int pad_l  = max(0, (W - rw) / 2);
        const int crop_l = max(0, (rw - W) / 2);

        Params p;
        p.rh = rh;             p.rw = rw;
        p.sy_off = crop_t - pad_t;
        p.sx_off = crop_l - pad_l;
        p.ym = ym;             p.xm = xm;
        p.hcm1 = hc - 1;       p.wcm1 = wc - 1;
        p.fy = (float)hc / (float)rh;
        p.fx = (float)wc / (float)rw;
        p.pad[0] = p.pad[1] = p.pad[2] = p.pad[3] = p.pad[4] = p.pad[5] = 0;
        params[blockIdx.x] = p;   // index == b*C + c
    }
}

// ---------------------------------------------------------------------------
// Kernel 4: bilinear focus. One thread per output element.
// Block = one (b,h) row slice: 8 w x 32 c. Params staged into LDS via TDM.
// ---------------------------------------------------------------------------
__global__ void focus_kernel(const float* __restrict__ x,
                             const Params* __restrict__ params,
                             float* __restrict__ out) {
    const int wblk = blockIdx.x & 63;
    const int h    = (blockIdx.x >> 6) & (H - 1);
    const int b    = blockIdx.x >> 15;          // 32768 blocks per batch
    const int c    = threadIdx.x & 31;
    const int w    = wblk * 8 + (threadIdx.x >> 5);

    __shared__ Params sp[C];                    // 2048 B

#if HAVE_TDM
    if (threadIdx.x < 32) {                     // wave 0 issues one TDM op
        const uint64_t ga = (uint64_t)(uintptr_t)(params + (size_t)b * C);
        const uint32_t la = (uint32_t)(uintptr_t)(&sp[0]);   // LDS byte offset
        // D# group 0: count=1, lds_addr, global_addr[56:0], type=2
        u32x4 g0 = { 1u, la, (uint32_t)ga,
                     (uint32_t)((ga >> 32) & 0x01FFFFFFull) | (2u << 30) };
        // D# group 1: data_size=4B; tensor_dim0=512, tensor_dim1=1;
        //             tile_dim0=512, tile_dim1=1; strides=512. (2048 B tile)
        i32x8 g1 = { 0x00020000, 0x02000000, 0x00010000, 0x02000000,
                     0x00000001, 0x00000200, 0x02000000, 0x00000000 };
        i32x4 gz4 = { 0, 0, 0, 0 };
        i32x8 gz8 = { 0, 0, 0, 0, 0, 0, 0, 0 };
        // 6-arg form (clang-23 / therock-10.0): (g0, g1, g2, g3, extra, cpol)
        __builtin_amdgcn_tensor_load_to_lds(g0, g1, gz4, gz4, gz8, 0);
        __builtin_amdgcn_s_wait_tensorcnt(0);   // s_wait_tensorcnt 0
    }
    __syncthreads();
#else
    if (threadIdx.x < 32) sp[threadIdx.x] = params[(size_t)b * C + threadIdx.x];
    __syncthreads();
#endif

    const Params& P = sp[c];
    const int ri = h + P.sy_off;
    const int rj = w + P.sx_off;

    float val = 0.0f;
    if (ri >= 0 && ri < P.rh && rj >= 0 && rj < P.rw) {
        const float sy = (float)ri * P.fy;      // TF1 legacy bilinear
        const float sx = (float)rj * P.fx;
        int y0 = min(max((int)sy, 0), P.hcm1);
        int x0 = min(max((int)sx, 0), P.wcm1);
        const int y1 = min(y0 + 1, P.hcm1);
        const int x1 = min(x0 + 1, P.wcm1);
        const float wy = fminf(fmaxf(sy - (float)y0, 0.0f), 1.0f);
        const float wx = fminf(fmaxf(sx - (float)x0, 0.0f), 1.0f);
        const int Y0 = min(P.ym + y0, H - 1);
        const int Y1 = min(P.ym + y1, H - 1);
        const int X0 = min(P.xm + x0, W - 1);
        const int X1 = min(P.xm + x1, W - 1);

        const float* img = x + ((size_t)b * H) * W * C;
        const float v00 = img[((size_t)Y0 * W + X0) * C + c];
        const float v01 = img[((size_t)Y0 * W + X1) * C + c];
        const float v10 = img[((size_t)Y1 * W + X0) * C + c];
        const float v11 = img[((size_t)Y1 * W + X1) * C + c];

        const float top = v00 * (1.0f - wx) + v01 * wx;
        const float bot = v10 * (1.0f - wx) + v11 * wx;
        val = top * (1.0f - wy) + bot * wy;
    }
    out[(((size_t)b * H + h) * W + w) * C + c] = val;
}

// ---------------------------------------------------------------------------
extern "C" void kernel_launch(void* const* d_in, const int* in_sizes, int n_in,
                              void* d_out, int out_size, void* d_ws, size_t ws_size,
                              hipStream_t stream) {
    (void)in_sizes; (void)n_in; (void)out_size; (void)ws_size;
    const float* x = (const float*)d_in[0];
    float* out = (float*)d_out;

    // Workspace: row_any (B*H u32, 8KB) | col_any (B*W u32, 8KB) | Params[B*C] (8KB)
    uint8_t* ws = (uint8_t*)d_ws;
    uint32_t* row_any = (uint32_t*)ws;
    uint32_t* col_any = row_any + (size_t)B * H;
    Params*   params  = (Params*)(ws + (size_t)(B * H + B * W) * 4);

    row_scan_kernel<<<B * H, 256, 0, stream>>>(x, row_any);
    col_scan_kernel<<<B * W, 256, 0, stream>>>(x, col_any);
    bbox_kernel<<<B * C, 512, 0, stream>>>(row_any, col_any, params);
    focus_kernel<<<B * (H * W * C / 256), 256, 0, stream>>>(x, params, out);
}